// EM_8864812499032
// MI455X (gfx1250) — compile-verified
//
#include <hip/hip_runtime.h>
#include <math.h>

// CDNA5 / gfx1250: wave32, WMMA f32 16x16x4.
typedef __attribute__((ext_vector_type(2))) float v2f;
typedef __attribute__((ext_vector_type(8))) float v8f;

#define MM   48          // mixture components
#define TT   80          // timesteps
#define NF   6           // final components
#define NEM  10          // EM iterations
#define NK   12          // K-steps of 4 (48 = 12*4)
#define LOG2PI 1.8378770664093453f

__device__ __constant__ int c_sel[3] = {29, 49, 79};

__global__ __launch_bounds__(256) void em_batch_kernel(
    const float* __restrict__ gLogits,   // (B,48)
    const float* __restrict__ gTraj,     // (B,48,80,2)
    const float* __restrict__ gCov,      // (B,48,80,2,2)
    float* __restrict__ outP,            // (B,6)
    float* __restrict__ outT,            // (B,6,80,2)
    float* __restrict__ outC)            // (B,6,80,2,2)
{
    const int b    = blockIdx.x;
    const int tid  = threadIdx.x;
    const int lane = tid & 31;
    const int wave = tid >> 5;

    __shared__ __align__(16) float sTraj[MM * 160];  // (m, t*2+d)
    __shared__ float sBuf[MM * MM];                  // R@t79, then adj
    __shared__ float sTsel[MM * 6];                  // (m, t'*2+d) selected times
    __shared__ float sPpad[16 * MM];                 // A matrix, rows 6..15 = 0
    __shared__ float sT6[NF * 160];                  // traj6
    __shared__ float sC6[NF * 320];                  // cov6 (t*4 + ij)
    __shared__ float sNum[NF * MM];
    __shared__ float sColInv[MM];
    __shared__ float sPrec[NF * 3 * 4];
    __shared__ float sLd[NF * 3];
    __shared__ float sPm[MM];                        // softmax(logits) over 48
    __shared__ float sP6[NF];
    __shared__ float sInvP6[NF];
    __shared__ float sScore[MM];
    __shared__ int   sIdx[NF];
    __shared__ unsigned sCnt;
    __shared__ unsigned sMin;
    __shared__ float sThr;

    const float* tb = gTraj + (size_t)b * (MM * 160);
    const float* cb = gCov  + (size_t)b * (MM * 320);
    const float* lb = gLogits + (size_t)b * MM;

    // Warm L2 with this batch's covariance block (global_prefetch_b8).
    __builtin_prefetch(cb + lane * 64, 0, 1);

    // ---- stage trajectories into LDS (float4 vectorized) ----
    for (int i = tid; i < (MM * 160) / 4; i += 256)
        ((float4*)sTraj)[i] = ((const float4*)tb)[i];
    // zero the padded A matrix once (rows 6..15 stay zero forever)
    for (int i = tid; i < 16 * MM; i += 256) sPpad[i] = 0.0f;

    // full softmax over 48 logits (thread 0; trivial)
    if (tid == 0) {
        float mx = -1e30f;
        for (int m = 0; m < MM; ++m) mx = fmaxf(mx, lb[m]);
        float s = 0.0f;
        for (int m = 0; m < MM; ++m) s += expf(lb[m] - mx);
        for (int m = 0; m < MM; ++m) sPm[m] = expf(lb[m] - mx) / s;
    }
    __syncthreads();

    // ---- selected-time points ----
    for (int e = tid; e < MM * 6; e += 256) {
        int m = e / 6, r = e % 6, tp = r >> 1, d = r & 1;
        sTsel[e] = sTraj[m * 160 + c_sel[tp] * 2 + d];
    }
    __syncthreads();

    // ---- pairwise distance at t=79 (for quantile threshold) ----
    for (int e = tid; e < MM * MM; e += 256) {
        int i = e / MM, j = e % MM;
        float xi = sTsel[i * 6 + 4], yi = sTsel[i * 6 + 5];
        float xj = sTsel[j * 6 + 4], yj = sTsel[j * 6 + 5];
        float d1i = xi * xi + yi * yi;
        float d1j = xj * xj + yj * yj;
        float dot = xi * xj + yi * yj;
        sBuf[e] = sqrtf(d1i + d1j - 2.0f * dot);
    }
    __syncthreads();

    // ---- exact 0.15-quantile (linear interp) via bitwise radix-select ----
    // Non-negative floats: IEEE bit patterns are order-isomorphic to values.
    {
        const float pos = 0.15f * (float)(MM * MM - 1);   // 345.45
        const int   kq  = (int)pos;                        // 345
        unsigned myBits[9];                                // 2304 = 256*9
        #pragma unroll
        for (int i = 0; i < 9; ++i)
            myBits[i] = __float_as_uint(sBuf[tid + 256 * i]);

        unsigned prefix = 0u;
        for (int bit = 31; bit >= 0; --bit) {
            unsigned cand = prefix | (1u << bit);
            if (tid == 0) sCnt = 0u;
            __syncthreads();
            int c = 0;
            #pragma unroll
            for (int i = 0; i < 9; ++i) c += (myBits[i] < cand) ? 1 : 0;
            for (int off = 16; off > 0; off >>= 1) c += __shfl_xor(c, off, 32);
            if (lane == 0) atomicAdd(&sCnt, (unsigned)c);
            __syncthreads();
            if (sCnt <= (unsigned)kq) prefix = cand;   // uniform decision
            __syncthreads();
        }
        const float v345 = __uint_as_float(prefix);

        // rank kq+1: duplicate of v345, or smallest element strictly greater
        if (tid == 0) { sCnt = 0u; sMin = 0xFFFFFFFFu; }
        __syncthreads();
        {
            int c = 0; unsigned mn = 0xFFFFFFFFu;
            #pragma unroll
            for (int i = 0; i < 9; ++i) {
                c += (myBits[i] <= prefix) ? 1 : 0;
                if (myBits[i] > prefix) mn = (myBits[i] < mn) ? myBits[i] : mn;
            }
            for (int off = 16; off > 0; off >>= 1) {
                c += __shfl_xor(c, off, 32);
                unsigned om = (unsigned)__shfl_xor((int)mn, off, 32);
                mn = (om < mn) ? om : mn;
            }
            if (lane == 0) { atomicAdd(&sCnt, (unsigned)c); atomicMin(&sMin, mn); }
        }
        __syncthreads();
        if (tid == 0) {
            float v346 = (sCnt >= (unsigned)(kq + 2)) ? v345 : __uint_as_float(sMin);
            sThr = v345 + (pos - (float)kq) * (v346 - v345);
        }
        __syncthreads();
    }

    // ---- adjacency: product over 3 selected times of (R <= thr) ----
    for (int e = tid; e < MM * MM; e += 256) {
        int i = e / MM, j = e % MM;
        float a = 1.0f;
        for (int tp = 0; tp < 3; ++tp) {
            float xi = sTsel[i * 6 + tp * 2], yi = sTsel[i * 6 + tp * 2 + 1];
            float xj = sTsel[j * 6 + tp * 2], yj = sTsel[j * 6 + tp * 2 + 1];
            float d1i = xi * xi + yi * yi;
            float d1j = xj * xj + yj * yj;
            float dot = xi * xj + yi * yj;
            float r = sqrtf(d1i + d1j - 2.0f * dot);
            a *= (r <= sThr) ? 1.0f : 0.0f;
        }
        sBuf[e] = a;   // safe: all threads hold their bits in registers
    }
    __syncthreads();

    // ---- base scores + greedy selection of 6 ----
    if (tid < MM) {
        float s = 0.0f;
        for (int j = 0; j < MM; ++j) s += sBuf[tid * MM + j] * sPm[j];
        sScore[tid] = s;
    }
    __syncthreads();
    if (tid == 0) {
        float worked[MM];
        for (int m = 0; m < MM; ++m) worked[m] = 1.0f;
        for (int it = 0; it < NF; ++it) {
            float best = -1e30f; int bi = 0;
            for (int m = 0; m < MM; ++m) {
                float sc = sScore[m] * worked[m];
                if (sc > best) { best = sc; bi = m; }
            }
            sIdx[it] = bi;
            for (int m = 0; m < MM; ++m) worked[m] *= (1.0f - sBuf[bi * MM + m]);
        }
        // probas6 init = softmax over selected logits
        float mx = -1e30f;
        for (int n = 0; n < NF; ++n) mx = fmaxf(mx, lb[sIdx[n]]);
        float s = 0.0f;
        for (int n = 0; n < NF; ++n) s += expf(lb[sIdx[n]] - mx);
        for (int n = 0; n < NF; ++n) sP6[n] = expf(lb[sIdx[n]] - mx) / s;
    }
    __syncthreads();

    // ---- gather initial traj6 / cov6 ----
    for (int e = tid; e < NF * 160; e += 256)
        sT6[e] = sTraj[sIdx[e / 160] * 160 + (e % 160)];
    for (int e = tid; e < NF * 320; e += 256)
        sC6[e] = cb[sIdx[e / 320] * 320 + (e % 320)];
    __syncthreads();

    const int k0sel = (lane < 16) ? 0 : 2;
    const int Mrow  = lane & 15;

    // ======================= EM iterations =======================
    for (int it = 0; it < NEM; ++it) {
        // A1: 2x2 inverses + logdets at selected times
        if (tid < NF * 3) {
            int n = tid / 3, tp = tid % 3;
            int base = n * 320 + c_sel[tp] * 4;
            float c00 = sC6[base], c01 = sC6[base + 1];
            float c10 = sC6[base + 2], c11 = sC6[base + 3];
            float det = c00 * c11 - c01 * c10;
            sLd[tid] = logf(det);
            float rd = 1.0f / det;
            sPrec[tid * 4 + 0] =  c11 * rd;
            sPrec[tid * 4 + 1] = -c01 * rd;
            sPrec[tid * 4 + 2] = -c10 * rd;
            sPrec[tid * 4 + 3] =  c00 * rd;
        }
        __syncthreads();

        // A2: responsibilities numerator
        for (int e = tid; e < NF * MM; e += 256) {
            int n = e / MM, m = e % MM;
            float s = 0.0f;
            for (int tp = 0; tp < 3; ++tp) {
                const float* p = &sPrec[(n * 3 + tp) * 4];
                int ts = c_sel[tp] * 2;
                float dx = sT6[n * 160 + ts]     - sTraj[m * 160 + ts];
                float dy = sT6[n * 160 + ts + 1] - sTraj[m * 160 + ts + 1];
                float q = dx * (p[0] * dx + p[1] * dy) + dy * (p[2] * dx + p[3] * dy);
                s += -LOG2PI - 0.5f * sLd[n * 3 + tp] - 0.5f * q;
            }
            sNum[e] = (expf(s) + 1e-8f) * sP6[n];
        }
        __syncthreads();

        // A3: reciprocal column sums
        if (tid < MM) {
            float c = 1e-8f;
            for (int n = 0; n < NF; ++n) c += sNum[n * MM + tid];
            sColInv[tid] = 1.0f / c;
        }
        __syncthreads();

        // A4: P = probas * coeff  (rows 0..5 of padded A matrix)
        for (int e = tid; e < NF * MM; e += 256) {
            int n = e / MM, m = e % MM;
            sPpad[n * MM + m] = sPm[m] * sNum[e] * sColInv[m];
        }
        __syncthreads();

        // A5: probas6 = row sums of P (+ hoisted reciprocal)
        if (tid < NF) {
            float s = 0.0f;
            for (int m = 0; m < MM; ++m) s += sPpad[tid * MM + m];
            sP6[tid] = s;
            sInvP6[tid] = 1.0f / s;
        }
        __syncthreads();

        // A-operand (P matrix) is identical for every tile: load once.
        v2f aArr[NK];
        #pragma unroll
        for (int kk = 0; kk < NK; ++kk) {
            int m0 = kk * 4 + k0sel;
            aArr[kk].x = sPpad[Mrow * MM + m0];
            aArr[kk].y = sPpad[Mrow * MM + m0 + 1];
        }

        // ---- Stage B: traj6 = P(6x48) x Traj(48x160) via v_wmma_f32_16x16x4_f32 ----
        for (int j = wave; j < 10; j += 8) {           // wave-uniform branch
            const int col = j * 16 + (lane & 15);
            v2f bArr[NK];
            #pragma unroll
            for (int kk = 0; kk < NK; ++kk) {          // issue all loads first
                int m0 = kk * 4 + k0sel;
                bArr[kk].x = sTraj[m0 * 160 + col];
                bArr[kk].y = sTraj[(m0 + 1) * 160 + col];
            }
            v8f acc = {};
            #pragma unroll
            for (int kk = 0; kk < NK; ++kk)
                acc = __builtin_amdgcn_wmma_f32_16x16x4_f32(
                          false, aArr[kk], false, bArr[kk], (short)0, acc, false, false);
            if (lane < 16) {
                for (int r = 0; r < NF; ++r)
                    sT6[r * 160 + col] = acc[r] * sInvP6[r];
            }
        }
        __syncthreads();

        // ---- Stage C: cov6 = P x (cov + x*x^T)(48x320) - probas6-normalized t6*t6^T ----
        for (int j = wave; j < 20; j += 8) {           // wave-uniform branch
            const int col = j * 16 + (lane & 15);
            const int t  = col >> 2, ij = col & 3;
            const int ii = ij >> 1,  jj = ij & 1;
            float c0[NK], c1[NK];
            #pragma unroll
            for (int kk = 0; kk < NK; ++kk) {          // issue all L2 loads up front
                int m0 = kk * 4 + k0sel;
                c0[kk] = cb[m0 * 320 + col];
                c1[kk] = cb[(m0 + 1) * 320 + col];
            }
            v2f bArr[NK];
            #pragma unroll
            for (int kk = 0; kk < NK; ++kk) {
                int m0 = kk * 4 + k0sel;
                bArr[kk].x = c0[kk]
                    + sTraj[m0 * 160 + t * 2 + ii] * sTraj[m0 * 160 + t * 2 + jj];
                bArr[kk].y = c1[kk]
                    + sTraj[(m0 + 1) * 160 + t * 2 + ii] * sTraj[(m0 + 1) * 160 + t * 2 + jj];
            }
            v8f acc = {};
            #pragma unroll
            for (int kk = 0; kk < NK; ++kk)
                acc = __builtin_amdgcn_wmma_f32_16x16x4_f32(
                          false, aArr[kk], false, bArr[kk], (short)0, acc, false, false);
            if (lane < 16) {
                for (int r = 0; r < NF; ++r)
                    sC6[r * 320 + col] = acc[r] * sInvP6[r]
                        - sT6[r * 160 + t * 2 + ii] * sT6[r * 160 + t * 2 + jj];
            }
        }
        __syncthreads();
    }

    // ---- write outputs ----
    if (tid < NF) outP[(size_t)b * NF + tid] = sP6[tid];
    for (int e = tid; e < NF * 160; e += 256) outT[(size_t)b * (NF * 160) + e] = sT6[e];
    for (int e = tid; e < NF * 320; e += 256) outC[(size_t)b * (NF * 320) + e] = sC6[e];
}

extern "C" void kernel_launch(void* const* d_in, const int* in_sizes, int n_in,
                              void* d_out, int out_size, void* d_ws, size_t ws_size,
                              hipStream_t stream) {
    (void)n_in; (void)out_size; (void)d_ws; (void)ws_size;
    const float* logits = (const float*)d_in[0];   // (B,48)
    const float* traj   = (const float*)d_in[1];   // (B,48,80,2)
    const float* cov    = (const float*)d_in[2];   // (B,48,80,2,2)
    const int B = in_sizes[0] / MM;

    float* outP = (float*)d_out;                   // (B,6)
    float* outT = outP + (size_t)B * NF;           // (B,6,80,2)
    float* outC = outT + (size_t)B * NF * 160;     // (B,6,80,2,2)

    em_batch_kernel<<<B, 256, 0, stream>>>(logits, traj, cov, outP, outT, outC);
}